// GenusSpeciesPoincare_20315195310439
// MI455X (gfx1250) — compile-verified
//
#include <hip/hip_runtime.h>
#include <math.h>

// ---------------------------------------------------------------------------
// CDNA5 (gfx1250) implementation.
// All matmuls use V_WMMA_F32_16X16X4_F32 (wave32, A:16x4 fp32 = v2f/lane,
// B:4x16 = v2f/lane, C/D:16x16 = v8f/lane).
// ---------------------------------------------------------------------------

typedef float v2f __attribute__((ext_vector_type(2)));
typedef float v8f __attribute__((ext_vector_type(8)));

__device__ __forceinline__ v8f wmma4(v2f a, v2f b, v8f c) {
  // (neg_a, A, neg_b, B, c_mod, C, reuse_a, reuse_b)
  return __builtin_amdgcn_wmma_f32_16x16x4_f32(false, a, false, b, (short)0, c,
                                               false, false);
}

// ---------------------------------------------------------------------------
// OUT[b][n] = bias[n]  (seed for split-K atomic accumulation). N power of 2.
// ---------------------------------------------------------------------------
__global__ void bias_init_kernel(float* __restrict__ out,
                                 const float* __restrict__ bias,
                                 int nMask, int total) {
  int i = blockIdx.x * blockDim.x + threadIdx.x;
  if (i < total) out[i] = bias[i & nMask];
}

// ---------------------------------------------------------------------------
// Exact (erf) GELU, in place.
// ---------------------------------------------------------------------------
__global__ void gelu_kernel(float* __restrict__ x, int total) {
  int i = blockIdx.x * blockDim.x + threadIdx.x;
  if (i < total) {
    float v = x[i];
    x[i] = 0.5f * v * (1.0f + erff(v * 0.70710678118654752440f));
  }
}

// ---------------------------------------------------------------------------
// C(64xN) += A(64xK) @ W(KxN), split-K over blockIdx.y, WMMA f32 16x16x4.
// One wave owns a 16-wide N tile and all 4 M tiles (B fragment reused 4x,
// so each weight element is loaded from HBM exactly once per split).
// Inner loop is fully strength-reduced: pointers bumped by constant strides,
// unconditional speculative prefetch 32 rows ahead of the weight stream.
// ---------------------------------------------------------------------------
__global__ void gemm_wmma_kernel(const float* __restrict__ A,
                                 const float* __restrict__ W,
                                 float* __restrict__ OUT,
                                 int K, int N, int kPer) {
  const int lane  = threadIdx.x & 31;
  const int wave  = threadIdx.x >> 5;
  const int tile  = blockIdx.x * 8 + wave;
  const int ntile = N >> 4;
  if (tile >= ntile) return;                 // wave-uniform

  const int    nBase = tile << 4;
  const int    col   = nBase + (lane & 15);
  const int    khalf = (lane >> 4) << 1;     // 0 or 2
  const int    r0    = lane & 15;
  const int    kBeg  = blockIdx.y * kPer;
  const int    kEnd  = min(K, kBeg + kPer);  // kPer multiple of 8
  const size_t ldw   = (size_t)N;
  const size_t lda   = (size_t)K;

  // Streaming pointers (constant-stride increments; no per-iter u64 muls).
  const float* wp = W + (size_t)(kBeg + khalf) * ldw + col;
  const float* ap = A + (size_t)r0 * lda + kBeg + khalf;
  const size_t a16 = 16 * lda, a32 = 32 * lda, a48 = 48 * lda;
  const size_t wstep = 8 * ldw;              // one k-unroll of weight rows
  const size_t wpf   = 32 * ldw;             // prefetch distance (rows)

  v8f acc0 = {}, acc1 = {}, acc2 = {}, acc3 = {};

  for (int kk = kBeg; kk < kEnd; kk += 8) {
    __builtin_prefetch(wp + wpf, 0, 1);      // speculative: OOB is dropped

    v2f b0, b1;
    b0.x = wp[0];        b0.y = wp[ldw];
    b1.x = wp[4 * ldw];  b1.y = wp[5 * ldw];

    v2f a00 = *(const v2f*)(ap);
    v2f a10 = *(const v2f*)(ap + a16);
    v2f a20 = *(const v2f*)(ap + a32);
    v2f a30 = *(const v2f*)(ap + a48);
    v2f a01 = *(const v2f*)(ap + 4);
    v2f a11 = *(const v2f*)(ap + a16 + 4);
    v2f a21 = *(const v2f*)(ap + a32 + 4);
    v2f a31 = *(const v2f*)(ap + a48 + 4);

    acc0 = wmma4(a00, b0, acc0);
    acc1 = wmma4(a10, b0, acc1);
    acc2 = wmma4(a20, b0, acc2);
    acc3 = wmma4(a30, b0, acc3);
    acc0 = wmma4(a01, b1, acc0);
    acc1 = wmma4(a11, b1, acc1);
    acc2 = wmma4(a21, b1, acc2);
    acc3 = wmma4(a31, b1, acc3);

    wp += wstep;
    ap += 8;
  }

  const int mOff = (lane >> 4) << 3;         // 0 or 8
  float* o0 = OUT + (size_t)( 0 + mOff) * ldw + col;
  float* o1 = OUT + (size_t)(16 + mOff) * ldw + col;
  float* o2 = OUT + (size_t)(32 + mOff) * ldw + col;
  float* o3 = OUT + (size_t)(48 + mOff) * ldw + col;
#pragma unroll
  for (int r = 0; r < 8; ++r) {
    atomicAdd(o0 + (size_t)r * ldw, acc0[r]);
    atomicAdd(o1 + (size_t)r * ldw, acc1[r]);
    atomicAdd(o2 + (size_t)r * ldw, acc2[r]);
    atomicAdd(o3 + (size_t)r * ldw, acc3[r]);
  }
}

// ---------------------------------------------------------------------------
// geoopt expmap0 per batch row (D = 256 = blockDim), also emits
// y2[b] = ||hyp_b||^2 = tanh^2(sqrt(c)*||u||)/c for the distance epilogue.
// ---------------------------------------------------------------------------
__global__ void expmap_kernel(const float* __restrict__ feat,
                              float* __restrict__ hyp,
                              float* __restrict__ y2) {
  __shared__ float red[256];
  const int b = blockIdx.x, t = threadIdx.x;
  const float v = feat[(size_t)b * 256 + t];
  red[t] = v * v;
  __syncthreads();
  for (int s = 128; s > 0; s >>= 1) {
    if (t < s) red[t] += red[t + s];
    __syncthreads();
  }
  const float sc = 1.22474487139158905f;     // sqrt(1.5)
  float n  = fmaxf(sqrtf(red[0]), 1e-15f);
  float th = tanhf(sc * n);
  hyp[(size_t)b * 256 + t] = v * (th / (sc * n));
  if (t == 0) y2[b] = (th * th) / 1.5f;
}

// ---------------------------------------------------------------------------
// x2[p] = ||protos[p]||^2
// ---------------------------------------------------------------------------
__global__ void sqnorm_kernel(const float* __restrict__ pts,
                              float* __restrict__ x2) {
  __shared__ float red[256];
  const int p = blockIdx.x, t = threadIdx.x;
  const float v = pts[(size_t)p * 256 + t];
  red[t] = v * v;
  __syncthreads();
  for (int s = 128; s > 0; s >>= 1) {
    if (t < s) red[t] += red[t + s];
    __syncthreads();
  }
  if (t == 0) x2[p] = red[0];
}

// ---------------------------------------------------------------------------
// Poincare distance logits via WMMA:
//   xy        = -<hyp_b, proto_p>         (64xP GEMM, K = 256)
//   ||num||^2 = a^2 x2 + 2ab xy + b^2 y2  (no (B,P,d) intermediate!)
// ---------------------------------------------------------------------------
__device__ __forceinline__ void dist_epilogue(v8f acc, int mBase, int mOff,
                                              int pcol, int P, float x2,
                                              const float* __restrict__ y2a,
                                              float* __restrict__ out) {
  const float Cc = 1.5f;
  const float sc = 1.22474487139158905f;     // sqrt(1.5)
#pragma unroll
  for (int r = 0; r < 8; ++r) {
    const int m = mBase + r + mOff;
    float y2  = y2a[m];
    float xy  = -acc[r];
    float aco = 1.0f + 2.0f * Cc * xy + Cc * y2;
    float bco = 1.0f - Cc * x2;
    float den = fmaxf(fabsf(1.0f + 2.0f * Cc * xy + Cc * Cc * x2 * y2), 1e-15f);
    float nm2 = aco * aco * x2 + 2.0f * aco * bco * xy + bco * bco * y2;
    float rr  = sc * sqrtf(fmaxf(nm2, 0.0f)) / den;
    rr = fminf(fmaxf(rr, 0.0f), 1.0f - 1e-5f);
    float dist = (2.0f / sc) * atanhf(rr);
    if (pcol < P) out[(size_t)m * P + pcol] = -dist * 10.0f;  // -dist / 0.1
  }
}

__global__ void dist_wmma_kernel(const float* __restrict__ hyp,
                                 const float* __restrict__ protos,
                                 const float* __restrict__ x2a,
                                 const float* __restrict__ y2a,
                                 float* __restrict__ out, int P) {
  const int lane  = threadIdx.x & 31;
  const int wave  = threadIdx.x >> 5;
  const int tile  = blockIdx.x * 8 + wave;
  const int ntile = (P + 15) >> 4;
  if (tile >= ntile) return;                 // wave-uniform

  const int nBase = tile << 4;
  const int pcol  = nBase + (lane & 15);
  const int p     = min(pcol, P - 1);        // clamp: keep EXEC all-1 for WMMA
  const int khalf = (lane >> 4) << 1;
  const int r0    = lane & 15;

  // Streaming pointers (protos^T columns == contiguous proto rows).
  const float* pp = protos + (size_t)p * 256 + khalf;
  const float* hp = hyp + (size_t)r0 * 256 + khalf;
  const size_t h16 = 16 * 256, h32 = 32 * 256, h48 = 48 * 256;

  v8f acc0 = {}, acc1 = {}, acc2 = {}, acc3 = {};

  for (int kk = 0; kk < 256; kk += 8) {
    v2f b0  = *(const v2f*)(pp);
    v2f b1  = *(const v2f*)(pp + 4);
    v2f a00 = *(const v2f*)(hp);
    v2f a10 = *(const v2f*)(hp + h16);
    v2f a20 = *(const v2f*)(hp + h32);
    v2f a30 = *(const v2f*)(hp + h48);
    v2f a01 = *(const v2f*)(hp + 4);
    v2f a11 = *(const v2f*)(hp + h16 + 4);
    v2f a21 = *(const v2f*)(hp + h32 + 4);
    v2f a31 = *(const v2f*)(hp + h48 + 4);

    acc0 = wmma4(a00, b0, acc0);
    acc1 = wmma4(a10, b0, acc1);
    acc2 = wmma4(a20, b0, acc2);
    acc3 = wmma4(a30, b0, acc3);
    acc0 = wmma4(a01, b1, acc0);
    acc1 = wmma4(a11, b1, acc1);
    acc2 = wmma4(a21, b1, acc2);
    acc3 = wmma4(a31, b1, acc3);

    pp += 8;
    hp += 8;
  }

  const float x2  = x2a[p];
  const int  mOff = (lane >> 4) << 3;
  dist_epilogue(acc0,  0, mOff, pcol, P, x2, y2a, out);
  dist_epilogue(acc1, 16, mOff, pcol, P, x2, y2a, out);
  dist_epilogue(acc2, 32, mOff, pcol, P, x2, y2a, out);
  dist_epilogue(acc3, 48, mOff, pcol, P, x2, y2a, out);
}

// ---------------------------------------------------------------------------
// Host orchestration.
// ---------------------------------------------------------------------------
extern "C" void kernel_launch(void* const* d_in, const int* in_sizes, int n_in,
                              void* d_out, int out_size, void* d_ws,
                              size_t ws_size, hipStream_t stream) {
  (void)in_sizes; (void)n_in; (void)out_size; (void)ws_size;

  const float* x   = (const float*)d_in[0];
  const float* pg  = (const float*)d_in[1];   // (500, 256)
  const float* ps  = (const float*)d_in[2];   // (5000, 256)
  const float* w1g = (const float*)d_in[3];  const float* b1g = (const float*)d_in[4];
  const float* w2g = (const float*)d_in[5];  const float* b2g = (const float*)d_in[6];
  const float* w3g = (const float*)d_in[7];  const float* b3g = (const float*)d_in[8];
  const float* w1s = (const float*)d_in[9];  const float* b1s = (const float*)d_in[10];
  const float* w2s = (const float*)d_in[11]; const float* b2s = (const float*)d_in[12];
  const float* w3s = (const float*)d_in[13]; const float* b3s = (const float*)d_in[14];
  float* out = (float*)d_out;

  const int B = 64, H = 4096, D = 256, PG = 500, PS = 5000;

  // Workspace carve-out (256B aligned slices).
  char* ws = (char*)d_ws;
  size_t off = 0;
  auto carve = [&](size_t bytes) -> float* {
    float* p = (float*)(ws + off);
    off += (bytes + 255) & ~(size_t)255;
    return p;
  };
  float* h1    = carve((size_t)B * H * 4);
  float* h2    = carve((size_t)B * H * 4);
  float* featg = carve((size_t)B * D * 4);
  float* feats = carve((size_t)B * D * 4);
  float* hypg  = carve((size_t)B * D * 4);
  float* hyps  = carve((size_t)B * D * 4);
  float* y2g   = carve((size_t)B * 4);
  float* y2s   = carve((size_t)B * 4);
  float* x2g   = carve((size_t)PG * 4);
  float* x2s   = carve((size_t)PS * 4);

  auto run_mlp = [&](const float* w1, const float* b1, const float* w2,
                     const float* b2, const float* w3, const float* b3,
                     float* feat, float* hyp, float* y2) {
    // layer 1: (64x768) @ (768x4096) + b1, GELU
    bias_init_kernel<<<(B * H + 255) / 256, 256, 0, stream>>>(h1, b1, H - 1, B * H);
    gemm_wmma_kernel<<<dim3((H / 16) / 8, 4), 256, 0, stream>>>(x, w1, h1, 768, H, 192);
    gelu_kernel<<<(B * H + 255) / 256, 256, 0, stream>>>(h1, B * H);
    // layer 2: (64x4096) @ (4096x4096) + b2, GELU
    bias_init_kernel<<<(B * H + 255) / 256, 256, 0, stream>>>(h2, b2, H - 1, B * H);
    gemm_wmma_kernel<<<dim3((H / 16) / 8, 8), 256, 0, stream>>>(h1, w2, h2, H, H, 512);
    gelu_kernel<<<(B * H + 255) / 256, 256, 0, stream>>>(h2, B * H);
    // layer 3: (64x4096) @ (4096x256) + b3
    bias_init_kernel<<<(B * D + 255) / 256, 256, 0, stream>>>(feat, b3, D - 1, B * D);
    gemm_wmma_kernel<<<dim3((D / 16) / 8, 16), 256, 0, stream>>>(h2, w3, feat, H, D, 256);
    // expmap0 + ||y||^2
    expmap_kernel<<<B, 256, 0, stream>>>(feat, hyp, y2);
  };

  run_mlp(w1g, b1g, w2g, b2g, w3g, b3g, featg, hypg, y2g);
  run_mlp(w1s, b1s, w2s, b2s, w3s, b3s, feats, hyps, y2s);

  sqnorm_kernel<<<PG, 256, 0, stream>>>(pg, x2g);
  sqnorm_kernel<<<PS, 256, 0, stream>>>(ps, x2s);

  {
    int ntg = (PG + 15) / 16;                 // 32 tiles
    int nts = (PS + 15) / 16;                 // 313 tiles
    dist_wmma_kernel<<<(ntg + 7) / 8, 256, 0, stream>>>(hypg, pg, x2g, y2g,
                                                        out, PG);
    dist_wmma_kernel<<<(nts + 7) / 8, 256, 0, stream>>>(hyps, ps, x2s, y2s,
                                                        out + (size_t)B * PG, PS);
  }
}